// Generalized_EM_4209067950484
// MI455X (gfx1250) — compile-verified
//
#include <hip/hip_runtime.h>
#include <math.h>

// ---------------------------------------------------------------------------
// Generalized EM graph solver for MI455X (gfx1250, wave32).
//
// Layouts (all in d_ws, float units):
//   f   : (N,16,3)  node-major FC features
//   w   : (N,16)    unscaled similarity weights
//   deg : (N)       unscaled degree
//   r,p,Ap,xs : (N,16) node-major CG vectors (one 64B line per node)
//   scalars: rs_it[11][16], pAp_it[10][16], musc = MU*scale
//   pb  : per-block reduction partials (deterministic two-stage reductions)
// ---------------------------------------------------------------------------

typedef float v2f __attribute__((ext_vector_type(2)));
typedef float v8f __attribute__((ext_vector_type(8)));

#define MU     0.5f
#define C_NORM 8.0f
#define LRELU  0.2f
#define EPS_CG 1e-12f
#define CG_ITERS 10

// ---------------------------------------------------------------------------
// S1: FC layer via V_WMMA_F32_16X16X4_F32. One wave per node n.
//   A (16x8, two K=4 tiles): row b = e = [x[b,n], emb[n,0..5], 0]
//   B (8x16): col j = fc_weight[j][:], cols 3..15 zero
//   C: bias[j] broadcast down rows
//   D[b][j] -> leakyrelu -> f[n][b][j]
// A-layout (ISA 7.12.2, 32-bit A 16x4): lane m (hi=0): K=0,1 ; lane m+16: K=2,3
// B-layout symmetric: lane j (hi=0): K=0,1 rows ; lane j+16: K=2,3 rows
// D-layout: VGPR v: lanes0-15 -> M=v, lanes16-31 -> M=v+8 ; N = lane&15
// ---------------------------------------------------------------------------
__global__ __launch_bounds__(256) void fc_wmma_kernel(
    const float* __restrict__ x, const float* __restrict__ emb,
    const float* __restrict__ W, const float* __restrict__ bias,
    float* __restrict__ f, int N) {
  int wid  = (blockIdx.x * blockDim.x + threadIdx.x) >> 5;
  int lane = threadIdx.x & 31;
  if (wid >= N) return;                      // wave-uniform: EXEC stays full
  int n  = wid;
  int m  = lane & 15;
  int hi = lane >> 4;

  const float* e = emb + (size_t)n * 6;
  float e0 = e[0], e1 = e[1], e2 = e[2], e3 = e[3], e4 = e[4], e5 = e[5];
  float xv = x[(size_t)m * N + n];           // batch b = m

  v2f A1, A2, B1, B2;
  A1[0] = hi ? e1 : xv;   A1[1] = hi ? e2 : e0;   // K = 0..3
  A2[0] = hi ? e5 : e3;   A2[1] = hi ? 0.f : e4;  // K = 4..7 (K=7 zero)

  float w0=0.f,w1=0.f,w2=0.f,w3=0.f,w4=0.f,w5=0.f,w6=0.f,bj=0.f;
  if (m < 3) {
    const float* wr = W + m * 7;
    w0=wr[0]; w1=wr[1]; w2=wr[2]; w3=wr[3]; w4=wr[4]; w5=wr[5]; w6=wr[6];
    bj = bias[m];
  }
  B1[0] = hi ? w2 : w0;   B1[1] = hi ? w3 : w1;
  B2[0] = hi ? w6 : w4;   B2[1] = hi ? 0.f : w5;

  v8f C;
  #pragma unroll
  for (int i = 0; i < 8; ++i) C[i] = bj;

  v8f D = __builtin_amdgcn_wmma_f32_16x16x4_f32(false, A2, false, B2,
                                                (short)0, C, false, false);
  D     = __builtin_amdgcn_wmma_f32_16x16x4_f32(false, A1, false, B1,
                                                (short)0, D, false, false);
  if (m < 3) {
    #pragma unroll
    for (int v = 0; v < 8; ++v) {
      float val = D[v];
      val = (val >= 0.f) ? val : LRELU * val;
      int b = v + 8 * hi;
      f[(size_t)n * 48 + b * 3 + m] = val;
    }
  }
}

// ---------------------------------------------------------------------------
// S2: w[n,k] = mean_b exp(-||f[b,n]-f[b,nl]||^2/(2*theta)); deg[n]=sum_k w.
// One thread per (n,k); node's 16 threads are one half-wave.
// Emits block partial of (sum deg^2 + sum w^2) for the F-norm scale.
// ---------------------------------------------------------------------------
__global__ __launch_bounds__(256) void wcalc_kernel(
    const int* __restrict__ nl, const float* __restrict__ f,
    const float* __restrict__ theta, float* __restrict__ w,
    float* __restrict__ deg, float* __restrict__ pb2, int N) {
  int t    = blockIdx.x * 256 + threadIdx.x;
  int lane = threadIdx.x & 31;
  int wv   = threadIdx.x >> 5;
  int n = t >> 4, k = t & 15;
  bool act = n < N;
  float wval = 0.f;
  if (act) {
    int j = nl[t];
    float inv2t = 0.5f / theta[0];
    const float* fn = f + (size_t)n * 48;
    const float* fj = f + (size_t)j * 48;
    float s = 0.f;
    #pragma unroll
    for (int b = 0; b < 16; ++b) {
      float d0 = fn[b*3+0] - fj[b*3+0];
      float d1 = fn[b*3+1] - fj[b*3+1];
      float d2 = fn[b*3+2] - fj[b*3+2];
      s += expf(-(d0*d0 + d1*d1 + d2*d2) * inv2t);
    }
    wval = s * 0.0625f;
    w[t] = wval;
  }
  // deg = sum over the 16 lanes of this node
  float dg = wval;
  for (int off = 1; off < 16; off <<= 1) dg += __shfl_xor(dg, off, 32);
  if (act && k == 0) deg[n] = dg;
  // norm contribution: each lane adds w^2, k==0 lane adds deg^2 once
  float contrib = wval * wval + ((k == 0) ? dg * dg : 0.f);
  for (int off = 1; off < 32; off <<= 1) contrib += __shfl_xor(contrib, off, 32);
  __shared__ float lds[8];
  if (lane == 0) lds[wv] = contrib;
  __syncthreads();
  if (threadIdx.x == 0) {
    float s = 0.f;
    #pragma unroll
    for (int i = 0; i < 8; ++i) s += lds[i];
    pb2[blockIdx.x] = s;
  }
}

// S3: musc = MU * C_NORM / sqrt(sum deg^2 + sum w^2)   (deterministic reduce)
__global__ __launch_bounds__(256) void nrm_musc_kernel(
    const float* __restrict__ pb2, int nb, float* __restrict__ musc) {
  __shared__ float lds[256];
  float s = 0.f;
  for (int i = threadIdx.x; i < nb; i += 256) s += pb2[i];
  lds[threadIdx.x] = s;
  __syncthreads();
  for (int off = 128; off; off >>= 1) {
    if ((int)threadIdx.x < off) lds[threadIdx.x] += lds[threadIdx.x + off];
    __syncthreads();
  }
  if (threadIdx.x == 0) musc[0] = MU * (C_NORM / sqrtf(lds[0]));
}

// I1: transpose x(B,N) -> r,p (N,16); xs=0; block partials of rs0[b]=sum x^2
__global__ __launch_bounds__(256) void init_kernel(
    const float* __restrict__ x, float* __restrict__ r, float* __restrict__ p,
    float* __restrict__ xs, float* __restrict__ pb, int N) {
  int n = blockIdx.x * 256 + threadIdx.x;
  bool act = n < N;
  float prt[16];
  #pragma unroll
  for (int b = 0; b < 16; ++b) {
    float v = act ? x[(size_t)b * N + n] : 0.f;
    if (act) {
      r[(size_t)n * 16 + b] = v;
      p[(size_t)n * 16 + b] = v;
      xs[(size_t)n * 16 + b] = 0.f;
    }
    prt[b] = v * v;
  }
  __shared__ float lds[128];
  int lane = threadIdx.x & 31, wv = threadIdx.x >> 5;
  #pragma unroll
  for (int b = 0; b < 16; ++b) {
    float s = prt[b];
    for (int off = 1; off < 32; off <<= 1) s += __shfl_xor(s, off, 32);
    if (lane == 0) lds[wv * 16 + b] = s;
  }
  __syncthreads();
  if (threadIdx.x < 16) {
    float s = 0.f;
    #pragma unroll
    for (int i = 0; i < 8; ++i) s += lds[i * 16 + threadIdx.x];
    pb[(size_t)blockIdx.x * 16 + threadIdx.x] = s;
  }
}

// reduce16: out[b] = sum_i pb[i*16+b], fixed order -> deterministic
__global__ __launch_bounds__(256) void reduce16_kernel(
    const float* __restrict__ pb, int nblocks, float* __restrict__ out) {
  int b = blockIdx.x;  // 0..15
  __shared__ float lds[256];
  float s = 0.f;
  for (int i = threadIdx.x; i < nblocks; i += 256) s += pb[(size_t)i * 16 + b];
  lds[threadIdx.x] = s;
  __syncthreads();
  for (int off = 128; off; off >>= 1) {
    if ((int)threadIdx.x < off) lds[threadIdx.x] += lds[threadIdx.x + off];
    __syncthreads();
  }
  if (threadIdx.x == 0) out[b] = lds[0];
}

// K1: Ap = p + musc*(deg*p - sum_k w*p[nbr]); block partials of p.Ap per b.
// Thread per (n, q) with q = batch-quad; each neighbor gather is one float4.
__global__ __launch_bounds__(256) void applyA_kernel(
    const int* __restrict__ nl, const float* __restrict__ w,
    const float* __restrict__ deg, const float* __restrict__ musc_p,
    const float* __restrict__ p, float* __restrict__ Ap,
    float* __restrict__ pb, int N) {
  int t = blockIdx.x * 256 + threadIdx.x;
  int n = t >> 2, q = t & 3;
  bool act = n < N;
  float r0 = 0.f, r1 = 0.f, r2 = 0.f, r3 = 0.f;
  if (act) {
    float musc = musc_p[0];
    const int4*   idx4 = (const int4*)(nl + (size_t)n * 16);
    const float4* w4   = (const float4*)(w + (size_t)n * 16);
    const float4  ps   = *(const float4*)(p + (size_t)n * 16 + q * 4);
    float dg = deg[n];
    float ax = 0.f, ay = 0.f, az = 0.f, aw = 0.f;
    #pragma unroll
    for (int kk = 0; kk < 4; ++kk) {
      int4   id = idx4[kk];
      float4 wf = w4[kk];
      float4 a;
      a = *(const float4*)(p + (size_t)id.x * 16 + q * 4);
      ax += wf.x * a.x; ay += wf.x * a.y; az += wf.x * a.z; aw += wf.x * a.w;
      a = *(const float4*)(p + (size_t)id.y * 16 + q * 4);
      ax += wf.y * a.x; ay += wf.y * a.y; az += wf.y * a.z; aw += wf.y * a.w;
      a = *(const float4*)(p + (size_t)id.z * 16 + q * 4);
      ax += wf.z * a.x; ay += wf.z * a.y; az += wf.z * a.z; aw += wf.z * a.w;
      a = *(const float4*)(p + (size_t)id.w * 16 + q * 4);
      ax += wf.w * a.x; ay += wf.w * a.y; az += wf.w * a.z; aw += wf.w * a.w;
    }
    float4 o;
    o.x = ps.x + musc * (dg * ps.x - ax);
    o.y = ps.y + musc * (dg * ps.y - ay);
    o.z = ps.z + musc * (dg * ps.z - az);
    o.w = ps.w + musc * (dg * ps.w - aw);
    *(float4*)(Ap + (size_t)n * 16 + q * 4) = o;
    r0 = ps.x * o.x; r1 = ps.y * o.y; r2 = ps.z * o.z; r3 = ps.w * o.w;
  }
  // reduce across lanes sharing q (q == lane&3): xor over {4,8,16}
  for (int off = 4; off < 32; off <<= 1) {
    r0 += __shfl_xor(r0, off, 32); r1 += __shfl_xor(r1, off, 32);
    r2 += __shfl_xor(r2, off, 32); r3 += __shfl_xor(r3, off, 32);
  }
  __shared__ float lds[128];
  int lane = threadIdx.x & 31, wv = threadIdx.x >> 5;
  if (lane < 4) {
    lds[wv * 16 + lane * 4 + 0] = r0;
    lds[wv * 16 + lane * 4 + 1] = r1;
    lds[wv * 16 + lane * 4 + 2] = r2;
    lds[wv * 16 + lane * 4 + 3] = r3;
  }
  __syncthreads();
  if (threadIdx.x < 16) {
    float s = 0.f;
    #pragma unroll
    for (int i = 0; i < 8; ++i) s += lds[i * 16 + threadIdx.x];
    pb[(size_t)blockIdx.x * 16 + threadIdx.x] = s;
  }
}

// K2: alpha=rs/(pAp+eps); xs+=alpha*p; r-=alpha*Ap; partials of r.r per b
__global__ __launch_bounds__(256) void updateXR_kernel(
    const float* __restrict__ rs, const float* __restrict__ pap,
    const float* __restrict__ p, const float* __restrict__ Ap,
    float* __restrict__ xs, float* __restrict__ r,
    float* __restrict__ pb, int N) {
  int t = blockIdx.x * 256 + threadIdx.x;
  int n = t >> 2, q = t & 3;
  bool act = n < N;
  float r0 = 0.f, r1 = 0.f, r2 = 0.f, r3 = 0.f;
  if (act) {
    int base = q * 4;
    float a0 = rs[base + 0] / (pap[base + 0] + EPS_CG);
    float a1 = rs[base + 1] / (pap[base + 1] + EPS_CG);
    float a2 = rs[base + 2] / (pap[base + 2] + EPS_CG);
    float a3 = rs[base + 3] / (pap[base + 3] + EPS_CG);
    size_t o = (size_t)n * 16 + base;
    float4 pv = *(const float4*)(p + o);
    float4 av = *(const float4*)(Ap + o);
    float4 xv = *(const float4*)(xs + o);
    float4 rv = *(const float4*)(r + o);
    xv.x += a0 * pv.x; xv.y += a1 * pv.y; xv.z += a2 * pv.z; xv.w += a3 * pv.w;
    rv.x -= a0 * av.x; rv.y -= a1 * av.y; rv.z -= a2 * av.z; rv.w -= a3 * av.w;
    *(float4*)(xs + o) = xv;
    *(float4*)(r + o) = rv;
    r0 = rv.x * rv.x; r1 = rv.y * rv.y; r2 = rv.z * rv.z; r3 = rv.w * rv.w;
  }
  for (int off = 4; off < 32; off <<= 1) {
    r0 += __shfl_xor(r0, off, 32); r1 += __shfl_xor(r1, off, 32);
    r2 += __shfl_xor(r2, off, 32); r3 += __shfl_xor(r3, off, 32);
  }
  __shared__ float lds[128];
  int lane = threadIdx.x & 31, wv = threadIdx.x >> 5;
  if (lane < 4) {
    lds[wv * 16 + lane * 4 + 0] = r0;
    lds[wv * 16 + lane * 4 + 1] = r1;
    lds[wv * 16 + lane * 4 + 2] = r2;
    lds[wv * 16 + lane * 4 + 3] = r3;
  }
  __syncthreads();
  if (threadIdx.x < 16) {
    float s = 0.f;
    #pragma unroll
    for (int i = 0; i < 8; ++i) s += lds[i * 16 + threadIdx.x];
    pb[(size_t)blockIdx.x * 16 + threadIdx.x] = s;
  }
}

// K3: beta=rs_new/(rs_old+eps); p = r + beta*p
__global__ __launch_bounds__(256) void pupdate_kernel(
    const float* __restrict__ rsn, const float* __restrict__ rso,
    const float* __restrict__ r, float* __restrict__ p, int N) {
  int t = blockIdx.x * 256 + threadIdx.x;
  int n = t >> 2, q = t & 3;
  if (n >= N) return;
  int base = q * 4;
  float b0 = rsn[base + 0] / (rso[base + 0] + EPS_CG);
  float b1 = rsn[base + 1] / (rso[base + 1] + EPS_CG);
  float b2 = rsn[base + 2] / (rso[base + 2] + EPS_CG);
  float b3 = rsn[base + 3] / (rso[base + 3] + EPS_CG);
  size_t o = (size_t)n * 16 + base;
  float4 rv = *(const float4*)(r + o);
  float4 pv = *(const float4*)(p + o);
  pv.x = rv.x + b0 * pv.x; pv.y = rv.y + b1 * pv.y;
  pv.z = rv.z + b2 * pv.z; pv.w = rv.w + b3 * pv.w;
  *(float4*)(p + o) = pv;
}

// F: transpose xs(N,16) -> out(B,N)
__global__ __launch_bounds__(256) void out_kernel(
    const float* __restrict__ xs, float* __restrict__ out, int N) {
  int n = blockIdx.x * 256 + threadIdx.x;
  int b = blockIdx.y;
  if (n < N) out[(size_t)b * N + n] = xs[(size_t)n * 16 + b];
}

// ---------------------------------------------------------------------------
extern "C" void kernel_launch(void* const* d_in, const int* in_sizes, int n_in,
                              void* d_out, int out_size, void* d_ws, size_t ws_size,
                              hipStream_t stream) {
  const float* x     = (const float*)d_in[0];   // (B,N)
  const int*   nl    = (const int*)d_in[1];     // (N,K)
  const float* emb   = (const float*)d_in[2];   // (N,E)
  const float* W     = (const float*)d_in[3];   // (F,E+1)
  const float* bias  = (const float*)d_in[4];   // (F)
  const float* theta = (const float*)d_in[5];   // scalar

  const int N = in_sizes[2] / 6;                // E = 6
  float* ws = (float*)d_ws;

  float* f_   = ws;                             // 48N
  float* w_   = ws + (size_t)48 * N;            // 16N
  float* deg_ = ws + (size_t)64 * N;            // N
  float* r_   = ws + (size_t)65 * N;            // 16N
  float* p_   = ws + (size_t)81 * N;            // 16N
  float* Ap_  = ws + (size_t)97 * N;            // 16N
  float* xs_  = ws + (size_t)113 * N;           // 16N
  float* sc   = ws + (size_t)129 * N;           // scalar block
  float* rs_it  = sc;                           // (CG_ITERS+1)*16
  float* pap_it = sc + 192;                     // CG_ITERS*16
  float* musc   = sc + 384;                     // 1
  float* pb     = sc + 512;                     // 16*4096 block partials
  float* pb2    = pb + 16 * 4096;               // 4096 scalar partials

  dim3 blk(256);
  int nbW = (N + 7) / 8;                 // one wave per node
  int nbK = (16 * N + 255) / 256;        // (n,k) threads
  int nbN = (N + 255) / 256;             // n threads
  int nbQ = (4 * N + 255) / 256;         // (n,q) threads

  // Setup: features (WMMA), weights, scale
  fc_wmma_kernel<<<nbW, blk, 0, stream>>>(x, emb, W, bias, f_, N);
  wcalc_kernel<<<nbK, blk, 0, stream>>>(nl, f_, theta, w_, deg_, pb2, N);
  nrm_musc_kernel<<<1, blk, 0, stream>>>(pb2, nbK, musc);

  // CG init: r = p = x, xs = 0, rs0 = sum r^2
  init_kernel<<<nbN, blk, 0, stream>>>(x, r_, p_, xs_, pb, N);
  reduce16_kernel<<<16, blk, 0, stream>>>(pb, nbN, rs_it);

  for (int it = 0; it < CG_ITERS; ++it) {
    applyA_kernel<<<nbQ, blk, 0, stream>>>(nl, w_, deg_, musc, p_, Ap_, pb, N);
    reduce16_kernel<<<16, blk, 0, stream>>>(pb, nbQ, pap_it + it * 16);
    updateXR_kernel<<<nbQ, blk, 0, stream>>>(rs_it + it * 16, pap_it + it * 16,
                                             p_, Ap_, xs_, r_, pb, N);
    reduce16_kernel<<<16, blk, 0, stream>>>(pb, nbQ, rs_it + (it + 1) * 16);
    if (it < CG_ITERS - 1)
      pupdate_kernel<<<nbQ, blk, 0, stream>>>(rs_it + (it + 1) * 16,
                                              rs_it + it * 16, r_, p_, N);
  }

  dim3 g((N + 255) / 256, 16);
  out_kernel<<<g, blk, 0, stream>>>(xs_, (float*)d_out, N);
}